// Decoder_77060303224971
// MI455X (gfx1250) — compile-verified
//
#include <hip/hip_runtime.h>
#include <hip/hip_bf16.h>
#include <math.h>

#define H 1024
#define L 128
#define V 50257

typedef __attribute__((ext_vector_type(2))) float v2f;
typedef __attribute__((ext_vector_type(8))) float v8f;

// ---------------------------------------------------------------------------
// Generic row-major mat-vec  y = act(W @ x + b)  via V_WMMA_F32_16X16X4_F32.
// One wave -> one 16-row tile. Vector x occupies row M=0 of the A fragment:
//   A (16x4 f32): lanes 0-15 hold M=lane {K=0,1}; lanes 16-31 hold M=lane-16 {K=2,3}
//   B (4x16 f32): VGPR0 lanes0-15=K0row / lanes16-31=K2row; VGPR1 = K1/K3 rows
//   D (16x16):    M=0 row lives in c[0] on lanes 0-15 (N = lane)
// A masking (only M=0 row nonzero) is done with a zeroed LDS page: lanes that
// must supply zeros read from zs[], lane 0/16 read from xs[] -> identical
// ds_load_b64 per lane, no VALU mask, no EXEC games.
// Inner loop processes 32 K per iteration in 3 phases (8 B-loads, 8 A-loads,
// 8 WMMAs on 2 accumulators) to keep ~8 global loads in flight per wave.
// K must be a multiple of 32 and <= 2048. ACT: 0=none, 1=relu.
// ---------------------------------------------------------------------------
template <int ACT>
__global__ void matvec_wmma_kernel(const float* __restrict__ W,
                                   const float* __restrict__ x,
                                   const float* __restrict__ bias,
                                   float* __restrict__ y,
                                   int N, int K) {
  __shared__ float xs[2048];
  __shared__ float zs[2048];     // zero page for A-row masking
  for (int i = threadIdx.x; i < K; i += blockDim.x) {
    xs[i] = x[i];
    zs[i] = 0.0f;
  }
  __syncthreads();

  const int lane          = threadIdx.x & 31;
  const int wavesPerBlock = blockDim.x >> 5;
  const int waveGlobal    = blockIdx.x * wavesPerBlock + (threadIdx.x >> 5);
  const int totalWaves    = gridDim.x * wavesPerBlock;
  const int ntiles        = (N + 15) >> 4;

  const int koff = (lane >= 16) ? 2 : 0;  // K pair this lane supplies
  const float* __restrict__ xbase = ((lane & 15) == 0) ? xs : zs;

  for (int tile = waveGlobal; tile < ntiles; tile += totalWaves) {
    const int n0   = tile << 4;
    const int row  = n0 + (lane & 15);
    const int rowc = (row < N) ? row : (N - 1);        // clamp for safe loads
    const float* __restrict__ Wr = W + (size_t)rowc * (size_t)K;

    v8f c0 = {0.f, 0.f, 0.f, 0.f, 0.f, 0.f, 0.f, 0.f};
    v8f c1 = {0.f, 0.f, 0.f, 0.f, 0.f, 0.f, 0.f, 0.f};

    for (int k0 = 0; k0 < K; k0 += 32) {
      v2f bv[8], av[8];
      // Phase 1: 8 streaming weight loads (single-use -> non-temporal)
      #pragma unroll
      for (int j = 0; j < 8; ++j)
        bv[j] = __builtin_nontemporal_load((const v2f*)(Wr + k0 + 4 * j + koff));
      // Phase 2: 8 A-fragment loads from LDS (broadcast / zero page)
      #pragma unroll
      for (int j = 0; j < 8; ++j)
        av[j] = *(const v2f*)(xbase + k0 + 4 * j + koff);
      // Phase 3: 8 WMMAs, alternating accumulators (no RAW chain)
      #pragma unroll
      for (int j = 0; j < 8; ++j) {
        if (j & 1)
          c1 = __builtin_amdgcn_wmma_f32_16x16x4_f32(false, av[j], false, bv[j],
                                                     (short)0, c1, false, false);
        else
          c0 = __builtin_amdgcn_wmma_f32_16x16x4_f32(false, av[j], false, bv[j],
                                                     (short)0, c0, false, false);
      }
    }

    if (lane < 16 && row < N) {
      float v = c0[0] + c1[0] + bias[row];   // D[0][lane] in VGPR0, lanes 0-15
      if (ACT == 1) v = fmaxf(v, 0.0f);
      y[row] = v;
    }
  }
}

// cat1 = [emb[x], h0]  (2048)
__global__ void prep_cat1_kernel(const int* __restrict__ x,
                                 const float* __restrict__ emb,
                                 const float* __restrict__ hidden,
                                 float* __restrict__ cat1) {
  int i = blockIdx.x * blockDim.x + threadIdx.x;
  if (i < 2 * H) {
    int idx = x[0];
    cat1[i] = (i < H) ? emb[(size_t)idx * H + i] : hidden[i - H];
  }
}

// softmax over 128 attention logits; write to ws and to d_out tail
__global__ void softmax128_kernel(const float* __restrict__ logits,
                                  float* __restrict__ w_ws,
                                  float* __restrict__ w_out) {
  __shared__ float red[L];
  int t = threadIdx.x;
  float v = logits[t];
  red[t] = v; __syncthreads();
  for (int s = L / 2; s > 0; s >>= 1) {
    if (t < s) red[t] = fmaxf(red[t], red[t + s]);
    __syncthreads();
  }
  float m = red[0]; __syncthreads();
  float e = expf(v - m);
  red[t] = e; __syncthreads();
  for (int s = L / 2; s > 0; s >>= 1) {
    if (t < s) red[t] += red[t + s];
    __syncthreads();
  }
  float w = e / red[0];
  w_ws[t]  = w;
  w_out[t] = w;
}

// context[h] = sum_l w[l] * enc[l][h]   (coalesced along h)
__global__ void context_kernel(const float* __restrict__ w,
                               const float* __restrict__ enc,
                               float* __restrict__ ctx) {
  __shared__ float wl[L];
  if (threadIdx.x < L) wl[threadIdx.x] = w[threadIdx.x];
  __syncthreads();
  int h = blockIdx.x * blockDim.x + threadIdx.x;
  if (h < H) {
    float acc = 0.0f;
    #pragma unroll 8
    for (int l = 0; l < L; ++l) acc += wl[l] * enc[l * H + h];
    ctx[h] = acc;
  }
}

// cat2 = [emb[x] (reuse cat1 low half), context]
__global__ void prep_cat2_kernel(const float* __restrict__ cat1,
                                 const float* __restrict__ ctx,
                                 float* __restrict__ cat2) {
  int i = blockIdx.x * blockDim.x + threadIdx.x;
  if (i < 2 * H) cat2[i] = (i < H) ? cat1[i] : ctx[i - H];
}

// GRU gates (PyTorch order r,z,n); biases already folded into gx/gh
__global__ void gru_combine_kernel(const float* __restrict__ gx,
                                   const float* __restrict__ gh,
                                   const float* __restrict__ h0,
                                   float* __restrict__ h_ws,
                                   float* __restrict__ h_out) {
  int i = blockIdx.x * blockDim.x + threadIdx.x;
  if (i < H) {
    float r = 1.0f / (1.0f + expf(-(gx[i] + gh[i])));
    float z = 1.0f / (1.0f + expf(-(gx[H + i] + gh[H + i])));
    float n = tanhf(gx[2 * H + i] + r * gh[2 * H + i]);
    float h = (1.0f - z) * n + z * h0[i];
    h_ws[i]  = h;
    h_out[i] = h;
  }
}

// single-block logsumexp over n logits
__global__ void lse_kernel(const float* __restrict__ logits, int n,
                           float* __restrict__ lse) {
  __shared__ float red[1024];
  int t = threadIdx.x;
  float m = -3.402823466e38f;
  for (int i = t; i < n; i += 1024) m = fmaxf(m, logits[i]);
  red[t] = m; __syncthreads();
  for (int s = 512; s > 0; s >>= 1) {
    if (t < s) red[t] = fmaxf(red[t], red[t + s]);
    __syncthreads();
  }
  m = red[0]; __syncthreads();
  float sum = 0.0f;
  for (int i = t; i < n; i += 1024) sum += expf(logits[i] - m);
  red[t] = sum; __syncthreads();
  for (int s = 512; s > 0; s >>= 1) {
    if (t < s) red[t] += red[t + s];
    __syncthreads();
  }
  if (t == 0) lse[0] = m + logf(red[0]);
}

// in-place: logp = logit - lse
__global__ void sub_lse_kernel(float* __restrict__ logp, int n,
                               const float* __restrict__ lse) {
  int i = blockIdx.x * blockDim.x + threadIdx.x;
  if (i < n) logp[i] = logp[i] - lse[0];
}

extern "C" void kernel_launch(void* const* d_in, const int* in_sizes, int n_in,
                              void* d_out, int out_size, void* d_ws, size_t ws_size,
                              hipStream_t stream) {
  const int*   x      = (const int*)  d_in[0];
  const float* hidden = (const float*)d_in[1];
  const float* enc    = (const float*)d_in[2];
  const float* emb    = (const float*)d_in[3];
  const float* attn_W = (const float*)d_in[4];
  const float* attn_b = (const float*)d_in[5];
  const float* comb_W = (const float*)d_in[6];
  const float* comb_b = (const float*)d_in[7];
  const float* W_ih   = (const float*)d_in[8];
  const float* W_hh   = (const float*)d_in[9];
  const float* b_ih   = (const float*)d_in[10];
  const float* b_hh   = (const float*)d_in[11];
  const float* out_W  = (const float*)d_in[12];
  const float* out_b  = (const float*)d_in[13];

  float* out      = (float*)d_out;
  float* out_logp = out;                 // V
  float* out_h    = out + V;             // H
  float* out_attn = out + V + H;         // L

  float* ws       = (float*)d_ws;
  float* ws_cat1  = ws;                  // 2048
  float* ws_alog  = ws + 2048;           // 128
  float* ws_aw    = ws + 2176;           // 128
  float* ws_ctx   = ws + 2304;           // 1024
  float* ws_cat2  = ws + 3328;           // 2048
  float* ws_rnn   = ws + 5376;           // 1024
  float* ws_gx    = ws + 6400;           // 3072
  float* ws_gh    = ws + 9472;           // 3072
  float* ws_h     = ws + 12544;          // 1024
  float* ws_lse   = ws + 13568;          // 1

  // 1. cat1 = [emb[x], h0]
  prep_cat1_kernel<<<8, 256, 0, stream>>>(x, emb, hidden, ws_cat1);
  // 2. attn logits = attn_W @ cat1 + attn_b   (128 x 2048 -> 8 tiles = 8 waves)
  matvec_wmma_kernel<0><<<1, 256, 0, stream>>>(attn_W, ws_cat1, attn_b, ws_alog, L, 2 * H);
  // 3. attention softmax
  softmax128_kernel<<<1, L, 0, stream>>>(ws_alog, ws_aw, out_attn);
  // 4. context = w^T @ enc
  context_kernel<<<4, 256, 0, stream>>>(ws_aw, enc, ws_ctx);
  // 5. cat2 = [emb[x], context]
  prep_cat2_kernel<<<8, 256, 0, stream>>>(ws_cat1, ws_ctx, ws_cat2);
  // 6. rnn_in = relu(comb_W @ cat2 + comb_b)  (1024 x 2048 -> 64 tiles)
  matvec_wmma_kernel<1><<<8, 256, 0, stream>>>(comb_W, ws_cat2, comb_b, ws_rnn, H, 2 * H);
  // 7. gx = W_ih @ rnn_in + b_ih              (3072 x 1024 -> 192 tiles)
  matvec_wmma_kernel<0><<<24, 256, 0, stream>>>(W_ih, ws_rnn, b_ih, ws_gx, 3 * H, H);
  // 8. gh = W_hh @ h0 + b_hh
  matvec_wmma_kernel<0><<<24, 256, 0, stream>>>(W_hh, hidden, b_hh, ws_gh, 3 * H, H);
  // 9. GRU combine -> h_new
  gru_combine_kernel<<<4, 256, 0, stream>>>(ws_gx, ws_gh, hidden, ws_h, out_h);
  // 10. logits = out_W @ h_new + out_b        (50257 x 1024 -> 3142 tiles)
  {
    int ntiles = (V + 15) / 16;
    int blocks = (ntiles + 7) / 8;         // 8 waves per 256-thread block
    matvec_wmma_kernel<0><<<blocks, 256, 0, stream>>>(out_W, ws_h, out_b, out_logp, V, H);
  }
  // 11. logsumexp over vocab
  lse_kernel<<<1, 1024, 0, stream>>>(out_logp, V, ws_lse);
  // 12. logp = logits - lse (in place)
  sub_lse_kernel<<<(V + 255) / 256, 256, 0, stream>>>(out_logp, V, ws_lse);
}